// SO3ConvolutionInvariants_53154515255873
// MI455X (gfx1250) — compile-verified
//
#include <hip/hip_runtime.h>

typedef float v2f __attribute__((ext_vector_type(2)));
typedef float v4f __attribute__((ext_vector_type(4)));
typedef float v8f __attribute__((ext_vector_type(8)));

// seg(d) = which l-segment dim d belongs to (0;1,1,1;2,2,2,2,2;3x7) == floor(sqrt(d))
__device__ __forceinline__ int seg_of(int d) {
    return (d == 0) ? 0 : (d < 4) ? 1 : (d < 9) ? 2 : 3;
}
// 1/sqrt(2l+1)
__device__ __forceinline__ float coef_of(int s) {
    return (s == 0) ? 1.0f
         : (s == 1) ? 0.57735026918962576f
         : (s == 2) ? 0.44721359549995794f
         :            0.37796447300922720f;
}

__device__ __forceinline__ v8f wmma_step(const v2f a, const v2f b, const v8f c) {
    // 8 args: (neg_a, A, neg_b, B, c_mod, C, reuse_a, reuse_b)
    return __builtin_amdgcn_wmma_f32_16x16x4_f32(false, a, false, b, (short)0, c,
                                                 false, false);
}

// Two 16-row tiles (32 rows) per wave per iteration.
//   D(16x16) += A_c(16x4) x B_c(4x16), c = 0..3  (K chunks of 4, permuted columns)
// K-permutation per chunk: c0=(0,1,4,5) c1=(2,3,6,7) c2=(8,9,12,13) c3=(10,11,14,15)
// => lo lanes load float4 cols 0-3 & 8-11 of their row, hi lanes cols 4-7 & 12-15:
//    two b128 loads per input per tile, every byte loaded exactly once, and the
//    B operand per chunk is the lane-uniform pair {q0.xy, q0.zw, q1.xy, q1.zw}.
// A_c[l][k] = M[col(c,k)][l] is a per-lane register constant (CG weights).
__global__ __launch_bounds__(256) void so3_inv_wmma_kernel(
    const float* __restrict__ x, const float* __restrict__ y,
    float* __restrict__ out, int npairs)
{
    const int lane = threadIdx.x & 31;
    const int L    = lane & 15;
    const int hi   = lane >> 4;
    const int wave   = (int)((blockIdx.x * blockDim.x + threadIdx.x) >> 5);
    const int nwaves = (int)((gridDim.x * blockDim.x) >> 5);

    // Loop-invariant A-matrix chunks under the permuted K ordering.
    v2f A[4];
#pragma unroll
    for (int c = 0; c < 4; ++c) {
        const int k0 = 8 * (c >> 1) + 4 * hi + 2 * (c & 1);
        const int s0 = seg_of(k0);
        const int s1 = seg_of(k0 + 1);
        v2f a;
        a.x = (s0 == L) ? coef_of(s0) : 0.0f;
        a.y = (s1 == L) ? coef_of(s1) : 0.0f;
        A[c] = a;
    }

    for (int p = wave; p < npairs; p += nwaves) {
        // float offsets; 16-byte aligned v4f accesses
        const size_t base0 = (size_t)p * 512 + (size_t)L * 16 + (size_t)(4 * hi);
        const size_t base1 = base0 + 256;

        v4f xa0 = __builtin_nontemporal_load((const v4f*)(x + base0));
        v4f xa1 = __builtin_nontemporal_load((const v4f*)(x + base0 + 8));
        v4f xb0 = __builtin_nontemporal_load((const v4f*)(x + base1));
        v4f xb1 = __builtin_nontemporal_load((const v4f*)(x + base1 + 8));
        v4f ya0 = __builtin_nontemporal_load((const v4f*)(y + base0));
        v4f ya1 = __builtin_nontemporal_load((const v4f*)(y + base0 + 8));
        v4f yb0 = __builtin_nontemporal_load((const v4f*)(y + base1));
        v4f yb1 = __builtin_nontemporal_load((const v4f*)(y + base1 + 8));

        const v4f pa0 = xa0 * ya0;   // tile0: chunk0 (xy), chunk1 (zw)
        const v4f pa1 = xa1 * ya1;   // tile0: chunk2 (xy), chunk3 (zw)
        const v4f pb0 = xb0 * yb0;   // tile1: chunk0, chunk1
        const v4f pb1 = xb1 * yb1;   // tile1: chunk2, chunk3

        v8f acc0 = {};
        v8f acc1 = {};
        // Interleave two independent accumulator chains for back-to-back issue.
        acc0 = wmma_step(A[0], (v2f){pa0.x, pa0.y}, acc0);
        acc1 = wmma_step(A[0], (v2f){pb0.x, pb0.y}, acc1);
        acc0 = wmma_step(A[1], (v2f){pa0.z, pa0.w}, acc0);
        acc1 = wmma_step(A[1], (v2f){pb0.z, pb0.w}, acc1);
        acc0 = wmma_step(A[2], (v2f){pa1.x, pa1.y}, acc0);
        acc1 = wmma_step(A[2], (v2f){pb1.x, pb1.y}, acc1);
        acc0 = wmma_step(A[3], (v2f){pa1.z, pa1.w}, acc0);
        acc1 = wmma_step(A[3], (v2f){pb1.z, pb1.w}, acc1);

        // D layout: VGPR j, lanes 0-15 => D[M=j][N=lane]; M-dim = l here, so
        // lanes 0-15 hold out[row_L][0..3] in acc[0..3] -> contiguous 256B stores.
        if (hi == 0) {
            v4f o0, o1;
            o0.x = acc0[0]; o0.y = acc0[1]; o0.z = acc0[2]; o0.w = acc0[3];
            o1.x = acc1[0]; o1.y = acc1[1]; o1.z = acc1[2]; o1.w = acc1[3];
            float* ob = out + (size_t)p * 128 + (size_t)L * 4;
            __builtin_nontemporal_store(o0, (v4f*)ob);
            __builtin_nontemporal_store(o1, (v4f*)(ob + 64));
        }
    }
}

// Scalar tail for rows % 32 (not hit for N_ROWS = 2,000,000 but kept for safety).
__global__ void so3_inv_tail_kernel(const float* __restrict__ x,
                                    const float* __restrict__ y,
                                    float* __restrict__ out,
                                    int start_row, int nrows)
{
    int r = start_row + (int)(blockIdx.x * blockDim.x + threadIdx.x);
    if (r >= nrows) return;
    float o0 = 0.f, o1 = 0.f, o2 = 0.f, o3 = 0.f;
#pragma unroll
    for (int d = 0; d < 16; ++d) {
        const float p = x[(size_t)r * 16 + d] * y[(size_t)r * 16 + d];
        const int s = seg_of(d);           // constant-folds per unrolled d
        if (s == 0) o0 += p;
        else if (s == 1) o1 += p;
        else if (s == 2) o2 += p;
        else o3 += p;
    }
    out[(size_t)r * 4 + 0] = o0 * coef_of(0);
    out[(size_t)r * 4 + 1] = o1 * coef_of(1);
    out[(size_t)r * 4 + 2] = o2 * coef_of(2);
    out[(size_t)r * 4 + 3] = o3 * coef_of(3);
}

extern "C" void kernel_launch(void* const* d_in, const int* in_sizes, int n_in,
                              void* d_out, int out_size, void* d_ws, size_t ws_size,
                              hipStream_t stream) {
    (void)n_in; (void)out_size; (void)d_ws; (void)ws_size;
    const float* x = (const float*)d_in[0];
    const float* y = (const float*)d_in[1];
    float* out = (float*)d_out;

    const int nrows  = in_sizes[0] / 16;
    const int npairs = nrows / 32;   // 32 rows (2 WMMA tiles) per wave-iteration

    if (npairs > 0) {
        // 8 waves per block; grid-stride so each wave handles a few pairs.
        long long blocksNeeded = ((long long)npairs + 7) / 8;
        int blocks = (blocksNeeded > 4096) ? 4096 : (int)blocksNeeded;
        so3_inv_wmma_kernel<<<blocks, 256, 0, stream>>>(x, y, out, npairs);
    }
    const int tail_start = npairs * 32;
    const int rem = nrows - tail_start;
    if (rem > 0) {
        int tb = (rem + 63) / 64;
        so3_inv_tail_kernel<<<tb, 64, 0, stream>>>(x, y, out, tail_start, nrows);
    }
}